// DifferentiableTVLayer_90520730731029
// MI455X (gfx1250) — compile-verified
//
#include <hip/hip_runtime.h>

// ---------------------------------------------------------------------------
// TV denoise (Chambolle-Pock), fully fused persistent kernel for gfx1250.
//   - 1 workgroup per image (B=8), 1024 threads = 32 wave32 waves.
//   - Full ubar image resident in LDS (256 x 257 f32 = 257KB of the 320KB WGP LDS,
//     pitch 257 -> conflict-free column access on 64 banks).
//   - px/py/u state in global (L2-resident: 192MB L2, whole problem ~10MB).
//   - Difference operators dx, dy, grad^T expressed as banded 256x256 matrix T
//     (-1 diag / +1 superdiag, with wx/wy zero-padding on the last row/col so the
//     padded formulation is EXACT), evaluated per 16x16 tile with chained
//     V_WMMA_F32_16X16X4_F32 ops; the superdiagonal block (single element) is a
//     halo row/col fix.
// ---------------------------------------------------------------------------

typedef float v2f __attribute__((ext_vector_type(2)));
typedef float v8f __attribute__((ext_vector_type(8)));

#define IMG   256
#define PITCH 257   // LDS row pitch in floats (conflict-free columns)

// T_d[r][c] : diagonal 16x16 block of T (-1 on diag, +1 on superdiag)
__device__ __forceinline__ float Td(int r, int c) {
    return (c == r) ? -1.0f : ((c == r + 1) ? 1.0f : 0.0f);
}
// T_d^T[r][c]
__device__ __forceinline__ float TdT(int r, int c) {
    return (r == c) ? -1.0f : ((r == c + 1) ? 1.0f : 0.0f);
}

__global__ __launch_bounds__(1024) void tv_chambolle_pock(
    const float* __restrict__ f_all,
    const float* __restrict__ lam_all,
    const int*   __restrict__ n_iter_ptr,
    float*       __restrict__ u_all,     // d_out doubles as u state
    float*       __restrict__ px_all,    // workspace, padded to 256x256 (row 255 == 0)
    float*       __restrict__ py_all)    // workspace, padded to 256x256 (col 255 == 0)
{
    extern __shared__ float sUbar[];     // [IMG][PITCH]

    const int b = blockIdx.x;
    const float* F = f_all   + (size_t)b * IMG * IMG;
    const float* L = lam_all + (size_t)b * IMG * IMG;
    float* U  = u_all  + (size_t)b * IMG * IMG;
    float* PX = px_all + (size_t)b * IMG * IMG;
    float* PY = py_all + (size_t)b * IMG * IMG;

    const int lane = threadIdx.x & 31;
    const int wave = threadIdx.x >> 5;
    const int hi   = lane >> 4;       // lane half (WMMA layout split)
    const int ln   = lane & 15;

    const float sigma  = 0.3535533905932738f;   // 1/sqrt(8)
    const float tau    = sigma;
    const float inv1pt = 1.0f / (1.0f + tau);
    const int n_iter   = n_iter_ptr[0];

    // ---------------- phase 0: u = ubar = f ; px = py = 0 ----------------
    #pragma unroll
    for (int i = 0; i < 8; ++i) {
        const int t  = wave * 8 + i;
        const int r0 = (t >> 4) * 16;
        const int c0 = (t & 15) * 16;
        #pragma unroll
        for (int v = 0; v < 8; ++v) {
            const int gr  = r0 + v + hi * 8;
            const int gc  = c0 + ln;
            const int idx = gr * IMG + gc;
            const float fv = F[idx];
            sUbar[gr * PITCH + gc] = fv;
            U[idx]  = fv;
            PX[idx] = 0.0f;
            PY[idx] = 0.0f;
        }
    }
    __syncthreads();

    for (int it = 0; it < n_iter; ++it) {
        // ============================ dual phase ============================
        #pragma unroll 1
        for (int i = 0; i < 8; ++i) {
            const int t  = wave * 8 + i;
            const int R  = t >> 4, C = t & 15;
            const int r0 = R * 16, c0 = C * 16;

            // ---- dx = T_d @ Ubar(R,C) via 4 chained WMMA f32 16x16x4 ----
            v8f ax = {};
            #pragma unroll
            for (int k = 0; k < 4; ++k) {
                const int kk = 4 * k + 2 * hi;
                v2f A, Bv;
                A[0]  = Td(ln, kk);
                A[1]  = Td(ln, kk + 1);
                Bv[0] = sUbar[(r0 + kk)     * PITCH + c0 + ln];
                Bv[1] = sUbar[(r0 + kk + 1) * PITCH + c0 + ln];
                ax = __builtin_amdgcn_wmma_f32_16x16x4_f32(
                        false, A, false, Bv, (short)0, ax, false, false);
            }
            // superdiagonal block T_u: row 15 += ubar[r0+16, :]
            {
                const int rn = (R < 15) ? (r0 + 16) : (IMG - 1);
                const float add = sUbar[rn * PITCH + c0 + ln];
                ax[7] += (hi && (R < 15)) ? add : 0.0f;   // M=15 lives in vgpr7 / hi lanes
            }
            // px = clip(px + sigma*dx, -wx, wx)  (wx row 255 padded to 0)
            #pragma unroll
            for (int v = 0; v < 8; ++v) {
                const int gr  = r0 + v + hi * 8;
                const int gc  = c0 + ln;
                const int idx = gr * IMG + gc;
                const int lr  = (gr < IMG - 1) ? (gr + 1) : (IMG - 1);
                const float wx = (gr < IMG - 1) ? L[lr * IMG + gc] : 0.0f;
                float p = PX[idx] + sigma * ax[v];
                p = fminf(fmaxf(p, -wx), wx);
                PX[idx] = p;
            }

            // ---- dy = Ubar(R,C) @ T_d^T via 4 chained WMMA ----
            v8f ay = {};
            #pragma unroll
            for (int k = 0; k < 4; ++k) {
                const int kk = 4 * k + 2 * hi;
                v2f A, Bv;
                A[0]  = sUbar[(r0 + ln) * PITCH + c0 + kk];
                A[1]  = sUbar[(r0 + ln) * PITCH + c0 + kk + 1];
                Bv[0] = TdT(kk, ln);
                Bv[1] = TdT(kk + 1, ln);
                ay = __builtin_amdgcn_wmma_f32_16x16x4_f32(
                        false, A, false, Bv, (short)0, ay, false, false);
            }
            // superdiagonal: col 15 += ubar[:, c0+16]
            {
                const int cn = (C < 15) ? (c0 + 16) : (IMG - 1);
                #pragma unroll
                for (int v = 0; v < 8; ++v) {
                    const float add = sUbar[(r0 + v + hi * 8) * PITCH + cn];
                    ay[v] += ((ln == 15) && (C < 15)) ? add : 0.0f;
                }
            }
            // py = clip(py + sigma*dy, -wy, wy)  (wy col 255 padded to 0)
            #pragma unroll
            for (int v = 0; v < 8; ++v) {
                const int gr  = r0 + v + hi * 8;
                const int gc  = c0 + ln;
                const int idx = gr * IMG + gc;
                const int lc  = (gc < IMG - 1) ? (gc + 1) : (IMG - 1);
                const float wy = (gc < IMG - 1) ? L[gr * IMG + lc] : 0.0f;
                float p = PY[idx] + sigma * ay[v];
                p = fminf(fmaxf(p, -wy), wy);
                PY[idx] = p;
            }
        }
        __threadfence();      // px/py halos cross waves through L2
        __syncthreads();

        // =========================== primal phase ===========================
        #pragma unroll 1
        for (int i = 0; i < 8; ++i) {
            const int t  = wave * 8 + i;
            const int R  = t >> 4, C = t & 15;
            const int r0 = R * 16, c0 = C * 16;

            // g = T_d^T @ Px(R,C)
            v8f g = {};
            #pragma unroll
            for (int k = 0; k < 4; ++k) {
                const int kk = 4 * k + 2 * hi;
                v2f A, Bv;
                A[0]  = TdT(ln, kk);
                A[1]  = TdT(ln, kk + 1);
                Bv[0] = PX[(r0 + kk)     * IMG + c0 + ln];
                Bv[1] = PX[(r0 + kk + 1) * IMG + c0 + ln];
                g = __builtin_amdgcn_wmma_f32_16x16x4_f32(
                        false, A, false, Bv, (short)0, g, false, false);
            }
            // subdiagonal block: row 0 += px[r0-1, :]
            {
                const int rp = (R > 0) ? (r0 - 1) : 0;
                const float add = PX[rp * IMG + c0 + ln];
                g[0] += ((hi == 0) && (R > 0)) ? add : 0.0f;  // M=0: vgpr0 / lo lanes
            }
            // g += Py(R,C) @ T_d
            #pragma unroll
            for (int k = 0; k < 4; ++k) {
                const int kk = 4 * k + 2 * hi;
                v2f A, Bv;
                A[0]  = PY[(r0 + ln) * IMG + c0 + kk];
                A[1]  = PY[(r0 + ln) * IMG + c0 + kk + 1];
                Bv[0] = Td(kk, ln);
                Bv[1] = Td(kk + 1, ln);
                g = __builtin_amdgcn_wmma_f32_16x16x4_f32(
                        false, A, false, Bv, (short)0, g, false, false);
            }
            // subdiagonal: col 0 += py[:, c0-1]
            {
                const int cp = (C > 0) ? (c0 - 1) : 0;
                #pragma unroll
                for (int v = 0; v < 8; ++v) {
                    const float add = PY[(r0 + v + hi * 8) * IMG + cp];
                    g[v] += ((ln == 0) && (C > 0)) ? add : 0.0f;
                }
            }
            // u_new = (u - tau*g + tau*f)/(1+tau) ; ubar = 2*u_new - u
            #pragma unroll
            for (int v = 0; v < 8; ++v) {
                const int gr  = r0 + v + hi * 8;
                const int gc  = c0 + ln;
                const int idx = gr * IMG + gc;
                const float uo = U[idx];
                const float un = (uo - tau * g[v] + tau * F[idx]) * inv1pt;
                U[idx] = un;
                sUbar[gr * PITCH + gc] = 2.0f * un - uo;
            }
        }
        __syncthreads();   // ubar handoff via LDS
    }
}

extern "C" void kernel_launch(void* const* d_in, const int* in_sizes, int n_in,
                              void* d_out, int out_size, void* d_ws, size_t ws_size,
                              hipStream_t stream) {
    const float* f   = (const float*)d_in[0];
    const float* lam = (const float*)d_in[1];
    const int*   nit = (const int*)d_in[2];
    float* u  = (float*)d_out;                       // 8*256*256 f32
    float* px = (float*)d_ws;                        // 2 MB
    float* py = px + (size_t)8 * IMG * IMG;          // 2 MB

    const size_t lds_bytes = (size_t)IMG * PITCH * sizeof(float);  // ~257 KB of 320 KB WGP LDS
    (void)hipFuncSetAttribute(reinterpret_cast<const void*>(tv_chambolle_pock),
                              hipFuncAttributeMaxDynamicSharedMemorySize,
                              (int)lds_bytes);

    tv_chambolle_pock<<<dim3(8), dim3(1024), lds_bytes, stream>>>(f, lam, nit, u, px, py);
}